// InferenceDAGMM_87823491269268
// MI455X (gfx1250) — compile-verified
//
#include <hip/hip_runtime.h>
#include <math.h>

typedef float v2f __attribute__((ext_vector_type(2)));
typedef float v8f __attribute__((ext_vector_type(8)));

#define KCOMP 8
#define DDIM 16

// DPP8 selector: lane i (within each 8-lane group) reads lane sel_i
#define DPP8SEL(s0,s1,s2,s3,s4,s5,s6,s7) \
  ((s0)|((s1)<<3)|((s2)<<6)|((s3)<<9)|((s4)<<12)|((s5)<<15)|((s6)<<18)|((s7)<<21))

// Butterfly sum over each 16-lane half of a wave32.
// xor1/2/4: DPP8 permutes (stay inside 8-lane groups);
// xor8 within a 16-lane row == row_ror:8 (DPP16 ctrl 0x128).
__device__ __forceinline__ float half_wave_sum16(float x) {
  int v = __float_as_int(x);
  x += __int_as_float(__builtin_amdgcn_mov_dpp8(v, DPP8SEL(1,0,3,2,5,4,7,6)));
  v = __float_as_int(x);
  x += __int_as_float(__builtin_amdgcn_mov_dpp8(v, DPP8SEL(2,3,0,1,6,7,4,5)));
  v = __float_as_int(x);
  x += __int_as_float(__builtin_amdgcn_mov_dpp8(v, DPP8SEL(4,5,6,7,0,1,2,3)));
  v = __float_as_int(x);
  x += __int_as_float(__builtin_amdgcn_update_dpp(v, v, 0x128 /*row_ror:8*/,
                                                  0xf, 0xf, false));
  return x;
}

// ---------------------------------------------------------------------------
// Setup kernel (one tiny block): per component k
//   L_k = cholesky(sigma_k); A_k = L_k^{-1}; v_k = A_k mu_k
//   lw_k = ln(pi_k) - 8*ln(2*pi) - sum(ln diag L_k)
// Workspace (floats): [0,2048) A_k row-major | [2048,2176) v_k | [2176,2184) lw
// ---------------------------------------------------------------------------
__global__ void dagmm_setup_kernel(const float* __restrict__ mu,
                                   const float* __restrict__ sigma,
                                   const float* __restrict__ pi,
                                   float* __restrict__ ws) {
  int k = threadIdx.x;
  if (k >= KCOMP) return;

  float L[DDIM][DDIM];
  for (int i = 0; i < DDIM; ++i)
    for (int j = 0; j < DDIM; ++j)
      L[i][j] = sigma[k * DDIM * DDIM + i * DDIM + j];

  for (int j = 0; j < DDIM; ++j) {
    float s = L[j][j];
    for (int p = 0; p < j; ++p) s -= L[j][p] * L[j][p];
    float d = sqrtf(s);
    L[j][j] = d;
    float inv = 1.0f / d;
    for (int i = j + 1; i < DDIM; ++i) {
      float t = L[i][j];
      for (int p = 0; p < j; ++p) t -= L[i][p] * L[j][p];
      L[i][j] = t * inv;
    }
  }
  float logdet = 0.0f;
  for (int j = 0; j < DDIM; ++j) logdet += logf(L[j][j]);

  float A[DDIM][DDIM];
  for (int i = 0; i < DDIM; ++i)
    for (int j = 0; j < DDIM; ++j) A[i][j] = 0.0f;
  for (int c = 0; c < DDIM; ++c) {
    for (int i = c; i < DDIM; ++i) {
      float s = (i == c) ? 1.0f : 0.0f;
      for (int p = c; p < i; ++p) s -= L[i][p] * A[p][c];
      A[i][c] = s / L[i][i];
    }
  }

  float* Aout = ws + k * DDIM * DDIM;
  for (int i = 0; i < DDIM; ++i)
    for (int j = 0; j < DDIM; ++j)
      Aout[i * DDIM + j] = A[i][j];

  float* vout = ws + KCOMP * DDIM * DDIM + k * DDIM;
  for (int i = 0; i < DDIM; ++i) {
    float s = 0.0f;
    for (int j = 0; j <= i; ++j) s += A[i][j] * mu[k * DDIM + j];
    vout[i] = s;
  }

  const float log_norm = -14.70301653127476f;  // -0.5 * 16 * ln(2*pi)
  ws[KCOMP * DDIM * DDIM + KCOMP * DDIM + k] = logf(pi[k]) + log_norm - logdet;
}

// ---------------------------------------------------------------------------
// Main kernel: one wave per 16-sample tile.
// D = Z_tile @ A_k^T + (-v_k)  via 4 chained V_WMMA_F32_16X16X4_F32 per k;
// maha via DPP butterfly; dens += exp2(lw2 - maha*0.5*log2e); out = -ln(dens).
// ---------------------------------------------------------------------------
__global__ void __launch_bounds__(256, 1)
dagmm_energy_kernel(const float* __restrict__ z,
                    const float* __restrict__ ws,
                    float* __restrict__ out,
                    int Bn, int ntiles) {
  const float* __restrict__ Ainv = ws;                                    // [8][16][16]
  const float* __restrict__ vv   = ws + KCOMP * DDIM * DDIM;              // [8][16]
  const float* __restrict__ lw   = ws + KCOMP * DDIM * DDIM + KCOMP*DDIM; // [8]

  const int lane = threadIdx.x & 31;
  const int wave = threadIdx.x >> 5;
  const int n    = lane & 15;   // N index for B/C/D operands
  const int hi   = lane >> 4;   // half-wave select

  const float LOG2E      = 1.4426950408889634f;
  const float NHALFLOG2E = -0.7213475204444817f;

  // B operands: bop[k][c].{x,y} = A_k[n][4c+2hi], A_k[n][4c+2hi+1]  (B = A_k^T)
  v2f   bop[KCOMP][4];
  float vneg[KCOMP];
  float lw2[KCOMP];   // forced to SGPRs (wave-uniform)
#pragma unroll
  for (int k = 0; k < KCOMP; ++k) {
#pragma unroll
    for (int c = 0; c < 4; ++c)
      bop[k][c] = *(const v2f*)&Ainv[k * 256 + n * 16 + 4 * c + 2 * hi];
    vneg[k] = -vv[k * DDIM + n];
    float l = lw[k] * LOG2E;  // log2-domain weight
    lw2[k] = __int_as_float(__builtin_amdgcn_readfirstlane(__float_as_int(l)));
  }

  const int wavesPerBlock = blockDim.x >> 5;
  const int wavesTotal    = gridDim.x * wavesPerBlock;
  const int wid           = blockIdx.x * wavesPerBlock + wave;

  for (int tile = wid; tile < ntiles; tile += wavesTotal) {
    const int base = tile * 16;

    // A operand: lane holds Z row (lane&15), cols 4c+2hi .. 4c+2hi+1
    int srow = base + n;
    if (srow >= Bn) srow = Bn - 1;  // tail safety (B divisible by 16 anyway)
    const float* zrow = z + (size_t)srow * DDIM;
    v2f a0 = *(const v2f*)&zrow[0  + 2 * hi];
    v2f a1 = *(const v2f*)&zrow[4  + 2 * hi];
    v2f a2 = *(const v2f*)&zrow[8  + 2 * hi];
    v2f a3 = *(const v2f*)&zrow[12 + 2 * hi];

    float dens[8];
#pragma unroll
    for (int r = 0; r < 8; ++r) dens[r] = 0.0f;

#pragma unroll
    for (int k = 0; k < KCOMP; ++k) {
      // C init = -v_k[n] broadcast -> D = A_k (z - mu_k) directly
      v8f acc;
#pragma unroll
      for (int r = 0; r < 8; ++r) acc[r] = vneg[k];

      acc = __builtin_amdgcn_wmma_f32_16x16x4_f32(false, a0, false, bop[k][0],
                                                  (short)0, acc, false, false);
      acc = __builtin_amdgcn_wmma_f32_16x16x4_f32(false, a1, false, bop[k][1],
                                                  (short)0, acc, false, false);
      acc = __builtin_amdgcn_wmma_f32_16x16x4_f32(false, a2, false, bop[k][2],
                                                  (short)0, acc, false, false);
      acc = __builtin_amdgcn_wmma_f32_16x16x4_f32(false, a3, false, bop[k][3],
                                                  (short)0, acc, false, false);

      // maha for sample (r + 8*hi): sum acc[r]^2 over the 16 lanes of each half
#pragma unroll
      for (int r = 0; r < 8; ++r) {
        float m = half_wave_sum16(acc[r] * acc[r]);
        dens[r] += exp2f(fmaf(m, NHALFLOG2E, lw2[k]));
      }
    }

#pragma unroll
    for (int r = 0; r < 8; ++r) {
      float e = -__logf(dens[r]);
      int sidx = base + r + 8 * hi;
      if (((lane & 15) == 0) && sidx < Bn) out[sidx] = e;
    }
  }
}

extern "C" void kernel_launch(void* const* d_in, const int* in_sizes, int n_in,
                              void* d_out, int out_size, void* d_ws, size_t ws_size,
                              hipStream_t stream) {
  const float* mu    = (const float*)d_in[0];  // [K,D]
  const float* sigma = (const float*)d_in[1];  // [K,D,D]
  const float* z     = (const float*)d_in[2];  // [B,D]
  const float* pi    = (const float*)d_in[3];  // [K]
  float* out = (float*)d_out;
  float* ws  = (float*)d_ws;

  const int Bn     = in_sizes[2] / DDIM;       // 524288
  const int ntiles = (Bn + 15) / 16;           // 32768

  hipLaunchKernelGGL(dagmm_setup_kernel, dim3(1), dim3(64), 0, stream,
                     mu, sigma, pi, ws);

  hipLaunchKernelGGL(dagmm_energy_kernel, dim3(1024), dim3(256), 0, stream,
                     z, ws, out, Bn, ntiles);
}